// BayesianSTDPAdaptive_61357902791398
// MI455X (gfx1250) — compile-verified
//
#include <hip/hip_runtime.h>
#include <math.h>

typedef float v2f __attribute__((ext_vector_type(2)));
typedef float v8f __attribute__((ext_vector_type(8)));

#define T_ITERS 400
#define TBATCH  10
#define O_DIM   256
#define I_DIM   512

// One wave (32 lanes) owns a 16(o) x 16(i) tile of the weight matrix and runs
// the full 400-step adaptive-STDP scan in registers. Each step's correlation
// tile corr = spk_tile^T (16x10) * psp_tile (10x16) is computed with three
// V_WMMA_F32_16X16X4_F32 ops (K padded 10 -> 12 with zero rows in LDS).
__global__ __launch_bounds__(32)
void stdp_weights_kernel(const float* __restrict__ psp,
                         const float* __restrict__ spk,
                         const float* __restrict__ w_in,
                         float* __restrict__ out)
{
    // 12 K-rows (rows 10,11 stay zero) so WMMA operand build needs no masks.
    __shared__ float spk_s[12 * 16];
    __shared__ float psp_s[12 * 16];
    __shared__ float tos_s[16];

    const int lane = threadIdx.x;      // 0..31
    const int row  = lane & 15;        // M index for A, N index for B
    const int khi  = lane >> 4;        // selects K-pair within a K=4 chunk
    const int i0   = blockIdx.x * 16;  // i tile base (0..511)
    const int o0   = blockIdx.y * 16;  // o tile base (0..255)

    // Zero the K-padding rows once (they are never written again).
    spk_s[160 + lane] = 0.0f;
    psp_s[160 + lane] = 0.0f;

    // Per-lane scan state: 8 weight elements, o = o0 + r + khi*8, i = i0 + row
    // (this is exactly the 16x16 f32 C/D VGPR layout of WMMA).
    float w[8], mu[8], m1[8], m2[8];
    const float m1w0 = logf(1.0f / (float)I_DIM);
    #pragma unroll
    for (int r = 0; r < 8; ++r) {
        const int o = o0 + r + khi * 8;
        w[r]  = w_in[o * I_DIM + i0 + row];
        mu[r] = 1.0f;
        m1[r] = m1w0;
        m2[r] = 2.0f + m1w0 * m1w0;
    }

    for (int t = 0; t < T_ITERS; ++t) {
        const int tb = t * TBATCH;

        // Stage the 10x16 spk (o-range) and psp (i-range) tiles into LDS.
        #pragma unroll
        for (int j = 0; j < 5; ++j) {
            const int f = lane + 32 * j;        // 0..159
            const int k = f >> 4;
            const int e = f & 15;
            spk_s[f] = spk[(tb + k) * O_DIM + o0 + e];
            psp_s[f] = psp[(tb + k) * I_DIM + i0 + e];
        }
        __syncthreads();

        // Per-o total output spikes for this chunk (needed by dw).
        if (lane < 16) {
            float s = 0.0f;
            #pragma unroll
            for (int k = 0; k < TBATCH; ++k) s += spk_s[k * 16 + lane];
            tos_s[lane] = s;
        }
        __syncthreads();

        // corr tile: 3 chained V_WMMA_F32_16X16X4_F32 (K = 0..11, rows 10,11 zero).
        // A layout (16x4 f32): lane -> M = lane%16; VGPR j holds K = (lane/16)*2 + j.
        // B layout mirrored:   lane -> N = lane%16; VGPR j holds K = (lane/16)*2 + j.
        v8f c = {};
        #pragma unroll
        for (int chunk = 0; chunk < 3; ++chunk) {
            const int kb = chunk * 4 + khi * 2;
            v2f a, b;
            a.x = spk_s[(kb    ) * 16 + row];
            a.y = spk_s[(kb + 1) * 16 + row];
            b.x = psp_s[(kb    ) * 16 + row];
            b.y = psp_s[(kb + 1) * 16 + row];
            c = __builtin_amdgcn_wmma_f32_16x16x4_f32(
                    /*neg_a=*/false, a, /*neg_b=*/false, b,
                    /*c_mod=*/(short)0, c, /*reuse_a=*/false, /*reuse_b=*/false);
        }

        // Elementwise adaptive-learning-rate update (C = 1.0 in reference).
        #pragma unroll
        for (int r = 0; r < 8; ++r) {
            const float tos = tos_s[r + khi * 8];
            const float dw  = c[r] * expf(-w[r]) - tos;
            const float wn  = w[r] + mu[r] * dw;
            const float om  = 1.0f - mu[r];
            const float m1n = mu[r] * wn + om * m1[r];
            const float m2n = mu[r] * wn * wn + om * m2[r];
            w[r]  = wn;
            m1[r] = m1n;
            m2[r] = m2n;
            mu[r] = (m2n - m1n * m1n) / (expf(-m1n) + 1.0f);
        }

        __syncthreads();  // LDS tiles reused next iteration
    }

    #pragma unroll
    for (int r = 0; r < 8; ++r) {
        const int o = o0 + r + khi * 8;
        out[o * I_DIM + i0 + row] = w[r];
    }
}

// Biases: 256 independent scans; tos_sum[t] is a block-wide reduction each step.
__global__ __launch_bounds__(256)
void stdp_bias_kernel(const float* __restrict__ spk,
                      const float* __restrict__ b_in,
                      float* __restrict__ out)
{
    __shared__ float red[256];
    const int o = threadIdx.x;

    float b  = b_in[o];
    float mu = 1.0f;
    float m1 = logf(1.0f / (float)O_DIM);
    float m2 = 2.0f + m1 * m1;

    for (int t = 0; t < T_ITERS; ++t) {
        const int tb = t * TBATCH;
        float tos = 0.0f;
        #pragma unroll
        for (int k = 0; k < TBATCH; ++k) tos += spk[(tb + k) * O_DIM + o];

        red[o] = tos;
        __syncthreads();
        #pragma unroll
        for (int s = 128; s > 0; s >>= 1) {
            if (o < s) red[o] += red[o + s];
            __syncthreads();
        }
        const float tsum = red[0];
        __syncthreads();

        const float db = (expf(-b) * tos - 1.0f) * tsum;
        b = b + mu * db;
        const float om = 1.0f - mu;
        m1 = mu * b + om * m1;
        m2 = mu * b * b + om * m2;
        mu = (m2 - m1 * m1) / (expf(-m1) + 1.0f);
    }

    out[o] = b;
}

extern "C" void kernel_launch(void* const* d_in, const int* in_sizes, int n_in,
                              void* d_out, int out_size, void* d_ws, size_t ws_size,
                              hipStream_t stream) {
    (void)in_sizes; (void)n_in; (void)out_size; (void)d_ws; (void)ws_size;
    const float* psp = (const float*)d_in[0];  // (4000, 512)
    const float* spk = (const float*)d_in[1];  // (4000, 256)
    const float* w   = (const float*)d_in[2];  // (256, 512)
    const float* b   = (const float*)d_in[3];  // (256,)
    float* out = (float*)d_out;                // [256*512 weights | 256 biases]

    dim3 grid(I_DIM / 16, O_DIM / 16);         // (32, 16) single-wave tiles
    stdp_weights_kernel<<<grid, 32, 0, stream>>>(psp, spk, w, out);
    stdp_bias_kernel<<<1, 256, 0, stream>>>(spk, b, out + O_DIM * I_DIM);
}